// PillarFeatureExtractor_5428838662336
// MI455X (gfx1250) — compile-verified
//
#include <hip/hip_runtime.h>

// ---------------------------------------------------------------------------
// PointPillars pillar-feature-extractor for gfx1250 (MI455X).
// One wave32 = one pillar: (32 x 76) x (76 x 64) GEMM via V_WMMA_F32_16X16X4_F32,
// fused BN + ReLU + max-over-points epilogue, scatter into the BEV grid.
// ---------------------------------------------------------------------------

typedef float v2f __attribute__((ext_vector_type(2)));
typedef float v8f __attribute__((ext_vector_type(8)));

#define M_PTS        32
#define CRAW         67          // 3 xyz + 64 feats per point in memory
#define KDIM         76          // 3 + 64 + 3 + 3 (+3 zero pad) == rows of W
#define KSTEPS       19          // 76 / 4
#define COUT_C       64
#define NX_C         432
#define NY_C         496
#define G_C          (NX_C * NY_C)
#define WAVES_PB     4
#define BLOCK_THR    (WAVES_PB * 32)
#define XS_STRIDE    77          // 13m mod 64 bijective -> conflict-free rows
#define XS_FLOATS    (M_PTS * XS_STRIDE)        // 2464
#define W_FLOATS     (KDIM * COUT_C)            // 4864
#define RAW_FLOATS   (M_PTS * CRAW)             // 2144
#define SMEM_FLOATS  (W_FLOATS + WAVES_PB * XS_FLOATS)
#define BN_EPS_C     0.001f

__global__ void pillar_wmma_kernel(const float* __restrict__ pillars,
                                   const int*   __restrict__ coords,
                                   const int*   __restrict__ num_points,
                                   const float* __restrict__ Wm,
                                   const float* __restrict__ gamma,
                                   const float* __restrict__ beta,
                                   const float* __restrict__ rmean,
                                   const float* __restrict__ rvar,
                                   float*       __restrict__ out,
                                   int n_pillars)
{
    extern __shared__ float smem[];
    float* lw = smem;                                   // W: 76 x 64
    const int tid  = threadIdx.x;
    const int wave = tid >> 5;
    const int lane = tid & 31;
    float* xs = smem + W_FLOATS + wave * XS_FLOATS;     // per-wave activation tile

    // --- cooperative W load (coalesced b128, L2-resident weights) ------------
    for (int i = tid; i < W_FLOATS / 4; i += BLOCK_THR)
        ((float4*)lw)[i] = ((const float4*)Wm)[i];
    __syncthreads();

    const int pillar = blockIdx.x * WAVES_PB + wave;
    if (pillar >= n_pillars) return;                    // wave-uniform: EXEC stays all-1s

    // --- Phase A: stage raw 32x67 tile into LDS with coalesced loads --------
    const float* src = pillars + (size_t)pillar * RAW_FLOATS;
    #pragma unroll
    for (int r = 0; r < 16; ++r) {
        int i = r * 32 + lane;
        ((float4*)xs)[i] = ((const float4*)src)[i];     // 2048 floats as b128
    }
    #pragma unroll
    for (int r = 0; r < 3; ++r) {
        int i = 2048 + r * 32 + lane;                   // 96-float tail
        xs[i] = src[i];
    }
    asm volatile("s_wait_dscnt 0" ::: "memory");        // cross-lane LDS visibility

    // --- Phase B: lane m owns point m; build masked 76-channel row ----------
    const int m = lane;
    float raw[CRAW];
    #pragma unroll
    for (int c = 0; c < CRAW; ++c) raw[c] = xs[m * CRAW + c];

    float sx = raw[0], sy = raw[1], sz = raw[2];        // mean over ALL 32 pts (as ref)
    #pragma unroll
    for (int off = 16; off >= 1; off >>= 1) {
        sx += __shfl_xor(sx, off);
        sy += __shfl_xor(sy, off);
        sz += __shfl_xor(sz, off);
    }
    const int   np   = num_points[pillar];
    const float inv  = 1.0f / (float)np;
    const float mxn = sx * inv, myn = sy * inv, mzn = sz * inv;

    const int bidx = coords[4 * pillar + 0];
    const int zi   = coords[4 * pillar + 1];
    const int yi   = coords[4 * pillar + 2];
    const int xi   = coords[4 * pillar + 3];
    const float cx = (float)xi * 0.16f + 0.08f;                  // PX/2 + PC_MIN.x
    const float cy = (float)yi * 0.16f + (0.08f - 39.68f);       // PY/2 + PC_MIN.y
    const float cz = (float)zi * 4.00f + (2.00f - 3.00f);        // PZ/2 + PC_MIN.z
    const float msk = (m < np) ? 1.0f : 0.0f;

    const int base = m * XS_STRIDE;
    #pragma unroll
    for (int c = 0; c < CRAW; ++c) xs[base + c] = raw[c] * msk;  // xyz + feats
    xs[base + 67] = (raw[0] - mxn) * msk;                        // cluster offsets
    xs[base + 68] = (raw[1] - myn) * msk;
    xs[base + 69] = (raw[2] - mzn) * msk;
    xs[base + 70] = (raw[0] - cx) * msk;                         // center offsets
    xs[base + 71] = (raw[1] - cy) * msk;
    xs[base + 72] = (raw[2] - cz) * msk;
    xs[base + 73] = 0.0f;                                        // K pad -> 76
    xs[base + 74] = 0.0f;
    xs[base + 75] = 0.0f;
    asm volatile("s_wait_dscnt 0" ::: "memory");

    // --- Phase C: GEMM via V_WMMA_F32_16X16X4_F32 ---------------------------
    // A 16x4 layout: lanes 0-15 M=0..15, VGPR0=K{0|2}, VGPR1=K{1|3} per half.
    const int half = lane >> 4;
    const int l15  = lane & 15;
    const int sidx = zi + yi * NX_C + xi;
    float* outbase = out + (size_t)bidx * COUT_C * G_C + sidx;

    #pragma unroll
    for (int nt = 0; nt < 4; ++nt) {
        v8f c0 = {};                                    // M-tile 0 (rows 0..15)
        v8f c1 = {};                                    // M-tile 1 (rows 16..31)
        const int col = nt * 16 + l15;
        #pragma unroll
        for (int ks = 0; ks < KSTEPS; ++ks) {
            const int k0 = ks * 4 + half * 2;
            v2f bfr, a0, a1;
            bfr.x = lw[ k0      * COUT_C + col];
            bfr.y = lw[(k0 + 1) * COUT_C + col];
            a0.x  = xs[ l15       * XS_STRIDE + k0];
            a0.y  = xs[ l15       * XS_STRIDE + k0 + 1];
            a1.x  = xs[(16 + l15) * XS_STRIDE + k0];
            a1.y  = xs[(16 + l15) * XS_STRIDE + k0 + 1];
            c0 = __builtin_amdgcn_wmma_f32_16x16x4_f32(
                     false, a0, false, bfr, (short)0, c0, false, false);
            c1 = __builtin_amdgcn_wmma_f32_16x16x4_f32(
                     false, a1, false, bfr, (short)0, c1, false, false);
        }
        // --- epilogue: BN + ReLU + max over 32 points -----------------------
        const int n   = nt * 16 + l15;
        const float sc = gamma[n] * rsqrtf(rvar[n] + BN_EPS_C);
        const float bi = beta[n] - rmean[n] * sc;
        float mx = 0.0f;                                // relu => max >= 0
        #pragma unroll
        for (int i = 0; i < 8; ++i) {
            mx = fmaxf(mx, c0[i] * sc + bi);
            mx = fmaxf(mx, c1[i] * sc + bi);
        }
        mx = fmaxf(mx, __shfl_xor(mx, 16));             // combine lane halves
        if (half == 0)
            outbase[(size_t)n * G_C] = mx;              // scatter into BEV grid
    }
}

extern "C" void kernel_launch(void* const* d_in, const int* in_sizes, int n_in,
                              void* d_out, int out_size, void* d_ws, size_t ws_size,
                              hipStream_t stream)
{
    (void)n_in; (void)d_ws; (void)ws_size;
    const float* pillars    = (const float*)d_in[0];
    const int*   coords     = (const int*)  d_in[1];
    const int*   num_points = (const int*)  d_in[2];
    const float* W          = (const float*)d_in[3];
    const float* gamma      = (const float*)d_in[4];
    const float* beta       = (const float*)d_in[5];
    const float* rmean      = (const float*)d_in[6];
    const float* rvar       = (const float*)d_in[7];
    float*       out        = (float*)d_out;

    const int n_pillars = in_sizes[2];                  // num_points count == N

    // Zero the 219 MB BEV canvas (stream memset is graph-capturable).
    hipMemsetAsync(out, 0, (size_t)out_size * sizeof(float), stream);

    const int    blocks = (n_pillars + WAVES_PB - 1) / WAVES_PB;
    const size_t smem   = SMEM_FLOATS * sizeof(float);  // 58880 B
    pillar_wmma_kernel<<<blocks, BLOCK_THR, smem, stream>>>(
        pillars, coords, num_points, W, gamma, beta, rmean, rvar, out, n_pillars);
}